// MahJongNetBatchedRevised_19198503813201
// MI455X (gfx1250) — compile-verified
//
#include <hip/hip_runtime.h>
#include <hip/hip_bf16.h>
#include <math.h>

// ---------------------------------------------------------------------------
// MahJongNetBatchedRevised on gfx1250 (MI455X).
// Memory-bound problem (~463 MB reads -> ~20us floor @ 23.3 TB/s).
// WMMA (V_WMMA_F32_16X16X4_F32) is used for the K=80 fan_pref reduction,
// which is free under the bandwidth roofline and exercises the CDNA5
// matrix pipe. Everything else is fused streaming f32.
// ---------------------------------------------------------------------------

typedef __attribute__((ext_vector_type(2))) float v2f;
typedef __attribute__((ext_vector_type(8))) float v8f;

// Kernel 1: multiplied_fan_pref[row] = sum_k fan_prep[row,k] * coeff[k]
// where coeff[k] = fan_coeff[k] * fan_coeff_multiplier[k], K = 80.
// One wave computes 16 rows via 20 chained V_WMMA_F32_16X16X4_F32.
//
// ISA layouts (CDNA5 ISA, 32-bit A 16x4):
//   A: lane L -> M = L&15 ; VGPR0 = A[M][k0 + 2*(L>>4)], VGPR1 = +1
//   B: lane L -> N = L&15 ; VGPR0 = B[k0 + 2*(L>>4)][N], VGPR1 = +1
//   D: lane L, VGPR i -> D[M = i + 8*(L>>4)][N = L&15]
// B columns are all set to coeff[k] -> every column of D carries the row sum.
__global__ void fan_reduce_wmma(const float* __restrict__ fan_prep,
                                const float* __restrict__ fan_coeff,
                                const float* __restrict__ fan_mult,
                                float* __restrict__ mfp,
                                int total_rows) {
    const int lane    = threadIdx.x & 31;
    const int wave    = blockIdx.x * (blockDim.x >> 5) + (threadIdx.x >> 5);
    const int rowbase = wave * 16;
    if (rowbase >= total_rows) return;  // wave-uniform guard (EXEC stays full)

    const int m     = lane & 15;
    const int khalf = (lane >> 4) << 1;  // 0 for lanes 0-15, 2 for lanes 16-31
    const float* __restrict__ rowp = fan_prep + (size_t)(rowbase + m) * 80;

    v8f c = {};
#pragma unroll
    for (int k0 = 0; k0 < 80; k0 += 4) {
        const int kk = k0 + khalf;
        v2f a, b;
        a.x = rowp[kk + 0];
        a.y = rowp[kk + 1];
        b.x = fan_coeff[kk + 0] * fan_mult[kk + 0];
        b.y = fan_coeff[kk + 1] * fan_mult[kk + 1];
        c = __builtin_amdgcn_wmma_f32_16x16x4_f32(
                /*neg_a=*/false, a, /*neg_b=*/false, b,
                /*c_mod=*/(short)0, c, /*reuse_a=*/false, /*reuse_b=*/false);
    }

    // All 16 columns identical: lane 0 holds M=0..7, lane 16 holds M=8..15.
    if (lane == 0) {
#pragma unroll
        for (int i = 0; i < 8; ++i) mfp[rowbase + i] = c[i];
    } else if (lane == 16) {
#pragma unroll
        for (int i = 0; i < 8; ++i) mfp[rowbase + 8 + i] = c[i];
    }
}

// Kernel 2: one block per batch element n, 320 threads (one per scenario t).
//  - threads 0..33 build prob_throw[n, tile] = dot(feat11, W) + b into LDS
//  - each thread streams its 34x7 tile_prep row, forms
//      term1 + null_filler + nan_to_num(cp_base*cp_coeff/hidden_ct)*cp_switch
//    takes the product over 34 tiles, scales by 100 * mfp.
__global__ void main_kernel(const float* __restrict__ meta,
                            const float* __restrict__ wall,
                            const float* __restrict__ tile_prep,
                            const float* __restrict__ W,
                            const float* __restrict__ bptr,
                            const float* __restrict__ mfp,
                            float* __restrict__ finalp) {
    const int n = blockIdx.x;
    const int t = threadIdx.x;  // 0..319

    __shared__ float s_prob[34];
    if (t < 34) {
        const float* mrow = meta + (size_t)n * 6;
        const float* wrow = wall + ((size_t)n * 34 + t) * 5;
        float s = bptr[0];
#pragma unroll
        for (int i = 0; i < 6; ++i) s = fmaf(mrow[i], W[i], s);
#pragma unroll
        for (int i = 0; i < 5; ++i) s = fmaf(wrow[i], W[6 + i], s);
        s_prob[t] = s;
    }
    __syncthreads();

    const size_t row = (size_t)n * 320 + t;
    const float* __restrict__ tp = tile_prep + row * (34 * 7);

    float prod = 1.0f;
#pragma unroll 2
    for (int tile = 0; tile < 34; ++tile) {
        const float* e = tp + tile * 7;
        const float c0 = e[0];  // self_draw_baseline
        const float c1 = e[1];  // need_count
        const float c2 = e[2];  // cp_coeff
        const float c3 = e[3];  // cp_base
        const float c4 = e[4];  // hidden_ct
        const float c5 = e[5];  // cp_switch
        const float c6 = e[6];  // null_filler

        const float term1 = c0 * powf(s_prob[tile], c1);

        float q = (c3 * c2) / c4;  // nan_to_num semantics
        if (isnan(q))       q = 0.0f;
        else if (isinf(q))  q = (q > 0.0f) ? 3.402823466e38f : -3.402823466e38f;

        prod *= (term1 + c6 + q * c5);
    }
    finalp[row] = prod * 100.0f * mfp[row];
}

// Kernel 3: one block per (n, action a), 64 threads.
//  tile_bound_winrate[n,a,t] = sum_{g<64} redundant[n, a*64+g, t] * final[n, a*64+g]
//  weighted_tile_prob[n,t]   = tbw[n,0,t] * tile_coeff[t]
//  max_tile_prob[n,a]        = a==0 ? sum_g final[n,g] : max_t tbw[n,a,t]
__global__ void reduce_kernel(const float* __restrict__ redundant,
                              const float* __restrict__ finalp,
                              const float* __restrict__ tile_coeff,
                              float* __restrict__ out,
                              int n_batch) {
    const int n   = blockIdx.x / 5;
    const int a   = blockIdx.x % 5;
    const int tid = threadIdx.x;  // 0..63

    __shared__ float sf[64];
    __shared__ float st[34];

    const size_t grp = (size_t)n * 320 + (size_t)a * 64;
    sf[tid] = finalp[grp + tid];
    __syncthreads();

    if (tid < 34) {
        const float* __restrict__ rp = redundant + grp * 34 + tid;
        float s = 0.0f;
#pragma unroll 4
        for (int g = 0; g < 64; ++g)
            s = fmaf(rp[(size_t)g * 34], sf[g], s);
        st[tid] = s;
        if (a == 0)
            out[(size_t)n_batch * 5 + (size_t)n * 34 + tid] = s * tile_coeff[tid];
    }
    __syncthreads();

    if (tid == 0) {
        float r;
        if (a == 0) {
            r = 0.0f;
            for (int g = 0; g < 64; ++g) r += sf[g];  // final_prob_ops[:,0]
        } else {
            r = st[0];
            for (int t = 1; t < 34; ++t) r = fmaxf(r, st[t]);
        }
        out[(size_t)n * 5 + a] = r;
    }
}

extern "C" void kernel_launch(void* const* d_in, const int* in_sizes, int n_in,
                              void* d_out, int out_size, void* d_ws, size_t ws_size,
                              hipStream_t stream) {
    const float* meta       = (const float*)d_in[0];   // (n,6)
    const float* wall       = (const float*)d_in[1];   // (n,34,5)
    const float* tile_prep  = (const float*)d_in[2];   // (n,320,34,7)
    const float* fan_prep   = (const float*)d_in[3];   // (n,320,80)
    const float* redundant  = (const float*)d_in[4];   // (n,320,34)
    // d_in[5] count_prep, d_in[6] chi_peng_count_remain: unused by reference
    const float* W          = (const float*)d_in[7];   // (11,)
    const float* b          = (const float*)d_in[8];   // scalar
    const float* fan_coeff  = (const float*)d_in[9];   // (80,)
    const float* fan_mult   = (const float*)d_in[10];  // (80,)
    const float* tile_coeff = (const float*)d_in[11];  // (34,)

    const int n = in_sizes[0] / 6;
    const int total_rows = n * 320;  // 327,680 for n=1024; multiple of 16

    float* mfp    = (float*)d_ws;                    // total_rows floats
    float* finalp = (float*)d_ws + (size_t)total_rows;  // total_rows floats

    // Kernel 1: WMMA fan reduction. 16 rows per wave, 4 waves per block.
    const int waves   = (total_rows + 15) / 16;
    const int blocks1 = (waves + 3) / 4;
    fan_reduce_wmma<<<blocks1, 128, 0, stream>>>(fan_prep, fan_coeff, fan_mult,
                                                 mfp, total_rows);

    // Kernel 2: fused prob_throw + term product + final_prob_throw.
    main_kernel<<<n, 320, 0, stream>>>(meta, wall, tile_prep, W, b, mfp, finalp);

    // Kernel 3: group reductions to both outputs.
    reduce_kernel<<<n * 5, 64, 0, stream>>>(redundant, finalp, tile_coeff,
                                            (float*)d_out, n);
}